// DirectEncoder_63917703299715
// MI455X (gfx1250) — compile-verified
//
#include <hip/hip_runtime.h>

// LIF current encoder (norse lif_current_encoder), SEQ_LENGTH=32.
//   v' = v + DT*TAU_MEM_INV*((V_LEAK - v) + x) = v + 0.1*(x - v)
//   z  = (v' >= V_TH) ? 1 : 0
//   v  = fired ? V_RESET : v'
// Output layout: zs[t][i], t in [0,32), i in [0,N). Pure store-bandwidth-bound:
// 201 MB out vs 6.3 MB in -> wide NT stores are the whole game on MI455X.

typedef __attribute__((ext_vector_type(4))) float v4f;

#define LIF_T 32
#define LIF_DECAY 0.9f      // 1 - DT*TAU_MEM_INV
#define LIF_GAIN 0.1f       // DT*TAU_MEM_INV
#define LIF_VTH 1.0f

__global__ __launch_bounds__(256) void lif_encode_vec4(const float* __restrict__ X,
                                                       float* __restrict__ Z,
                                                       int N) {
    const int i4 = blockIdx.x * blockDim.x + threadIdx.x;   // float4 index
    const int nvec = N >> 2;
    if (i4 >= nvec) return;

    // One-shot streaming read of the input current (never reused -> NT).
    v4f x = __builtin_nontemporal_load((const v4f*)X + i4);

    v4f xs;
    v4f v;
#pragma unroll
    for (int c = 0; c < 4; ++c) {
        xs[c] = LIF_GAIN * x[c];
        v[c]  = 0.0f;
    }

    // Fully unrolled time loop: 32 back-to-back global_store_b128 th:NT,
    // deep STOREcnt pipelining; serial fma chain has 4-wide ILP per thread.
#pragma unroll
    for (int t = 0; t < LIF_T; ++t) {
        v4f z;
#pragma unroll
        for (int c = 0; c < 4; ++c) {
            float vv   = fmaf(LIF_DECAY, v[c], xs[c]);  // == v + 0.1*(x - v)
            bool fired = (vv >= LIF_VTH);
            z[c] = fired ? 1.0f : 0.0f;
            v[c] = fired ? 0.0f : vv;
        }
        // Output is write-once, never re-read: bypass L2 residency (TH=NT).
        __builtin_nontemporal_store(z, (v4f*)(Z + (size_t)t * (size_t)N) + i4);
    }
}

// Safety fallback if N % 4 != 0 (not hit for the 8*3*256*256 shape).
__global__ __launch_bounds__(256) void lif_encode_scalar(const float* __restrict__ X,
                                                         float* __restrict__ Z,
                                                         int N) {
    const int i = blockIdx.x * blockDim.x + threadIdx.x;
    if (i >= N) return;
    const float xs = LIF_GAIN * X[i];
    float v = 0.0f;
#pragma unroll
    for (int t = 0; t < LIF_T; ++t) {
        float vv   = fmaf(LIF_DECAY, v, xs);
        bool fired = (vv >= LIF_VTH);
        float z    = fired ? 1.0f : 0.0f;
        v          = fired ? 0.0f : vv;
        __builtin_nontemporal_store(z, Z + (size_t)t * (size_t)N + i);
    }
}

extern "C" void kernel_launch(void* const* d_in, const int* in_sizes, int n_in,
                              void* d_out, int out_size, void* d_ws, size_t ws_size,
                              hipStream_t stream) {
    (void)n_in; (void)out_size; (void)d_ws; (void)ws_size;
    const float* X = (const float*)d_in[0];
    float*       Z = (float*)d_out;
    const int    N = in_sizes[0];

    const int block = 256;  // 8 wave32 per block
    if ((N & 3) == 0) {
        const int nvec = N >> 2;
        const int grid = (nvec + block - 1) / block;
        lif_encode_vec4<<<grid, block, 0, stream>>>(X, Z, N);
    } else {
        const int grid = (N + block - 1) / block;
        lif_encode_scalar<<<grid, block, 0, stream>>>(X, Z, N);
    }
}